// RGCN_20418274525635
// MI455X (gfx1250) — compile-verified
//
#include <hip/hip_runtime.h>
#include <cstdint>
#include <cstddef>

// ---------------------------------------------------------------------------
// RGCN (4-relation GraphConv, norm='both') for MI455X / gfx1250.
//   out = tanh( sum_r  Dr_in^-1/2 * scatter_dst( Dr_out^-1/2[src] * (X@W[r])[src] ) )
// GEMM uses V_WMMA_F32_16X16X4_F32 (exact fp32, matches reference precision).
// Scatter uses hardware global_atomic_add_f32; accumulator is L2-resident
// (25.6 MB << 192 MB L2), so the HBM floor is indices + x (~0.5 GB/call).
// ---------------------------------------------------------------------------

typedef __attribute__((ext_vector_type(2))) float v2f;
typedef __attribute__((ext_vector_type(8))) float v8f;

#define R_REL 4
#define K_IN  256
#define F_OUT 64

// ---- 1) zero counters + accumulator ---------------------------------------
__global__ __launch_bounds__(256)
void rgcn_zero_kernel(unsigned* __restrict__ cnt, long long ncnt,
                      float* __restrict__ agg, long long nagg)
{
    const long long stride = (long long)gridDim.x * blockDim.x;
    for (long long i = (long long)blockIdx.x * blockDim.x + threadIdx.x;
         i < ncnt || i < nagg; i += stride) {
        if (i < ncnt) cnt[i] = 0u;
        if (i < nagg) agg[i] = 0.0f;
    }
}

// ---- 2) degree histograms (out-degree of src, in-degree of dst) -----------
__global__ __launch_bounds__(256)
void rgcn_degree_kernel(const int* __restrict__ ei,   // [R,2,E] int32
                        unsigned* __restrict__ cS,    // [R,N]
                        unsigned* __restrict__ cD,    // [R,N]
                        int E, int N)
{
    const long long t = (long long)blockIdx.x * blockDim.x + threadIdx.x;
    const long long total = (long long)R_REL * E;
    if (t >= total) return;
    const int r = (int)(t / E);
    const int e = (int)(t - (long long)r * E);
    const int* base = ei + (size_t)r * 2 * E;
    const int s = base[e];
    const int d = base[(size_t)E + e];
    atomicAdd(&cS[(size_t)r * N + s], 1u);
    atomicAdd(&cD[(size_t)r * N + d], 1u);
}

// ---- 3) counts -> rsqrt(max(count,1)) -------------------------------------
__global__ __launch_bounds__(256)
void rgcn_norm_kernel(const unsigned* __restrict__ cnt,
                      float* __restrict__ nrm, long long total)
{
    const long long i = (long long)blockIdx.x * blockDim.x + threadIdx.x;
    if (i >= total) return;
    const unsigned c = cnt[i];
    const float dg = (float)(c > 0u ? c : 1u);
    nrm[i] = 1.0f / sqrtf(dg);
}

// ---- 4a) h = (X @ W[r]) * norm_src[:,None]  via fp32 WMMA -----------------
// Block = 256 threads = 8 waves. W[r] (256x64 = 64 KB) staged in LDS.
// Each wave owns a 16x16 output tile: wave>>2 selects row-half (2x16 rows per
// block), wave&3 selects 16-col tile. K-loop: 64 x V_WMMA_F32_16X16X4_F32.
//
// Fragment layouts (ISA 7.12.2, wave32):
//  A (16x4 fp32):  lane = {row = lane&15}; VGPR0 = K[2*(lane>>4)], VGPR1 = K+1
//  B (4x16 fp32):  lane = {col = lane&15}; VGPR0 = K[2*(lane>>4)], VGPR1 = K+1
//  C/D (16x16):    acc[v]: row = v + 8*(lane>>4), col = lane&15
//
// OOB rows: A is loaded from a clamped (valid) row pointer without zeroing --
// a bogus A row only corrupts the same output row, which the store guards.
__global__ __launch_bounds__(256)
void rgcn_gemm_kernel(const float* __restrict__ x,      // [N,256]
                      const float* __restrict__ Wr,     // [256,64]
                      const float* __restrict__ normS,  // [N]
                      float* __restrict__ h,            // [N,64]
                      int N)
{
    __shared__ float lW[K_IN * F_OUT];                  // 64 KB (of 320 KB WGP LDS)

    const int tid = threadIdx.x;
    {   // coalesced 128-bit staging of W into LDS
        const float4* __restrict__ Wv = (const float4*)Wr;
        float4* lWv = (float4*)lW;
        #pragma unroll
        for (int i = tid; i < (K_IN * F_OUT) / 4; i += 256) lWv[i] = Wv[i];
    }
    __syncthreads();

    const int wave     = tid >> 5;                      // 0..7
    const int lane     = tid & 31;
    const int laneHalf = lane >> 4;                     // 0/1
    const int lm       = lane & 15;
    const int rowTile  = blockIdx.x * 32 + (wave >> 2) * 16;
    const int colTile  = (wave & 3) * 16;

    const int aRow = rowTile + lm;
    // clamp OOB rows to a valid address; no zero-select needed (see above)
    const float* __restrict__ xrow = x + (size_t)(aRow < N ? aRow : 0) * K_IN;

    v8f acc = {};
    #pragma unroll 8
    for (int kk = 0; kk < K_IN; kk += 4) {
        const int ka = kk + laneHalf * 2;
        const float2 av = *(const float2*)(xrow + ka);  // 8B-aligned -> b64 load
        v2f a; a.x = av.x; a.y = av.y;
        v2f b;
        b.x = lW[(size_t)ka       * F_OUT + colTile + lm];
        b.y = lW[(size_t)(ka + 1) * F_OUT + colTile + lm];
        acc = __builtin_amdgcn_wmma_f32_16x16x4_f32(
                  false, a, false, b, (short)0, acc, false, false);
    }

    #pragma unroll
    for (int v = 0; v < 8; ++v) {
        const int row = rowTile + v + 8 * laneHalf;
        if (row < N) {
            const float ns = normS[row];
            h[(size_t)row * F_OUT + colTile + lm] = acc[v] * ns;
        }
    }
}

// ---- 4b) agg[dst] += h[src] * norm_dst[dst]  (one wave per edge) ----------
// e is wave-uniform (readfirstlane) so src/dst/norm_dst lower to scalar
// s_load via the constant cache; only the h-row gather and the 2 fp32
// atomics per lane hit the vector memory path.
__global__ __launch_bounds__(256)
void rgcn_scatter_kernel(const float* __restrict__ h,       // [N,64]
                         const int* __restrict__ srcdst,    // [2,E] (src|dst)
                         const float* __restrict__ nD,      // [N]
                         float* __restrict__ agg,           // [N,64]
                         int E)
{
    const int wave = __builtin_amdgcn_readfirstlane((int)threadIdx.x >> 5);
    const int lane = (int)threadIdx.x & 31;              // lane -> 2 columns
    const long long e = (long long)blockIdx.x * 8 + wave;
    if (e >= E) return;
    const int s = srcdst[e];                             // wave-uniform
    const int d = srcdst[(size_t)E + e];                 // wave-uniform
    const float nd = nD[d];                              // wave-uniform
    const float2 v = *(const float2*)(h + (size_t)s * F_OUT + 2 * lane);
    float* ap = agg + (size_t)d * F_OUT + 2 * lane;
    unsafeAtomicAdd(ap,     v.x * nd);                   // global_atomic_add_f32
    unsafeAtomicAdd(ap + 1, v.y * nd);
}

// ---- 5) out = tanh(agg), 128-bit vectorized -------------------------------
__global__ __launch_bounds__(256)
void rgcn_tanh_kernel(const float* __restrict__ agg, float* __restrict__ out,
                      long long n4)
{
    const long long i = (long long)blockIdx.x * blockDim.x + threadIdx.x;
    if (i >= n4) return;
    const float4 v = ((const float4*)agg)[i];
    float4 o;
    o.x = tanhf(v.x); o.y = tanhf(v.y); o.z = tanhf(v.z); o.w = tanhf(v.w);
    ((float4*)out)[i] = o;
}

// ---------------------------------------------------------------------------
extern "C" void kernel_launch(void* const* d_in, const int* in_sizes, int n_in,
                              void* d_out, int out_size, void* d_ws, size_t ws_size,
                              hipStream_t stream)
{
    (void)n_in; (void)out_size; (void)ws_size;

    const float* x  = (const float*)d_in[0];            // [N,256] fp32
    const float* W  = (const float*)d_in[1];            // [4,256,64] fp32
    const int*   ei = (const int*)d_in[2];              // [4,2,E] int32
    float*       out = (float*)d_out;                   // [N,64] fp32

    const int N = in_sizes[0] / K_IN;                   // 100000
    const int E = in_sizes[2] / (R_REL * 2);            // 1600000

    // workspace carve-out (256B aligned): counts | norms | h | agg  (~57 MB)
    char* ws = (char*)d_ws;
    size_t off = 0;
    auto take = [&](size_t bytes) -> void* {
        void* p = ws + off;
        off = (off + bytes + 255) & ~(size_t)255;
        return p;
    };
    unsigned* cnt = (unsigned*)take((size_t)2 * R_REL * N * sizeof(unsigned));
    float*    nrm = (float*)   take((size_t)2 * R_REL * N * sizeof(float));
    float*    h   = (float*)   take((size_t)N * F_OUT * sizeof(float));
    float*    agg = (float*)   take((size_t)N * F_OUT * sizeof(float));
    unsigned* cS = cnt;
    unsigned* cD = cnt + (size_t)R_REL * N;
    float*    nS = nrm;
    float*    nD = nrm + (size_t)R_REL * N;

    const long long ncnt = (long long)2 * R_REL * N;    // 800k
    const long long nagg = (long long)N * F_OUT;        // 6.4M

    rgcn_zero_kernel<<<4096, 256, 0, stream>>>(cnt, ncnt, agg, nagg);

    const long long totE = (long long)R_REL * E;
    rgcn_degree_kernel<<<(int)((totE + 255) / 256), 256, 0, stream>>>(ei, cS, cD, E, N);

    rgcn_norm_kernel<<<(int)((ncnt + 255) / 256), 256, 0, stream>>>(cnt, nrm, ncnt);

    const int gemmBlocks = (N + 31) / 32;               // 3125
    const int scatBlocks = (E + 7) / 8;                 // 200000 (8 waves/block)
    for (int r = 0; r < R_REL; ++r) {
        rgcn_gemm_kernel<<<gemmBlocks, 256, 0, stream>>>(
            x, W + (size_t)r * K_IN * F_OUT, nS + (size_t)r * N, h, N);
        rgcn_scatter_kernel<<<scatBlocks, 256, 0, stream>>>(
            h, ei + (size_t)r * 2 * E, nD + (size_t)r * N, agg, E);
    }

    const long long n4 = nagg / 4;
    rgcn_tanh_kernel<<<(int)((n4 + 255) / 256), 256, 0, stream>>>(agg, out, n4);
}